// LengthRegulator_13254269075841
// MI455X (gfx1250) — compile-verified
//
#include <hip/hip_runtime.h>
#include <hip/hip_bf16.h>

// Problem constants (from reference): B=16, C=256, T=1024, DMAX=8
#define B_    16
#define C_    256
#define T_    1024
#define MAXF_ 8192  // T_*DMAX

// Gather-kernel tiling
#define TPB 256   // threads per block (8 waves, wave32)
#define TP  1024  // frames per block = TPB*4 (16B stores)
#define CC  32    // channels per block

// Native vector types (amdgcn builtins want real vector types, not HIP's
// float4/int4 class wrappers).
typedef int   v4i __attribute__((vector_size(16)));
typedef float v4f __attribute__((vector_size(16)));

#if defined(__has_builtin)
#if __has_builtin(__builtin_amdgcn_global_load_async_to_lds_b128) && \
    __has_builtin(__builtin_amdgcn_s_wait_asynccnt)
#define HAVE_ASYNC_LDS 1
#endif
#endif
#ifndef HAVE_ASYNC_LDS
#define HAVE_ASYNC_LDS 0
#endif

// -----------------------------------------------------------------------------
// Kernel A: per-batch inclusive scan of clamped durations, frame_lengths, then
// searchsorted(cum, p, side='right') for all 8192 frame positions via binary
// search in LDS. Two-level wave32 shuffle scan (2 barriers instead of 20).
// One block per batch, 1024 threads = 32 waves.
// -----------------------------------------------------------------------------
__global__ __launch_bounds__(1024) void lr_scan_tok_kernel(
    const int* __restrict__ duration,  // [B][T]
    int* __restrict__ tok,             // [B][MAXF] (workspace)
    int* __restrict__ flen,            // [B]       (workspace)
    float* __restrict__ out_flen)      // [B]       (tail of d_out, as float)
{
  __shared__ int s[T_];     // inclusive cumsum
  __shared__ int wsum[32];  // per-wave totals
  const int b    = blockIdx.x;
  const int t    = threadIdx.x;
  const int lane = t & 31;
  const int wave = t >> 5;

  int d = duration[b * T_ + t];
  if (d < 0) d = 0;  // jnp.maximum(duration, 0)

  // Wave-level inclusive scan (wave32 shuffles).
  int v = d;
  #pragma unroll
  for (int off = 1; off < 32; off <<= 1) {
    int u = __shfl_up(v, off, 32);
    if (lane >= off) v += u;
  }
  if (lane == 31) wsum[wave] = v;
  __syncthreads();

  // Wave 0 scans the 32 wave totals.
  if (wave == 0) {
    int w = wsum[lane];
    #pragma unroll
    for (int off = 1; off < 32; off <<= 1) {
      int u = __shfl_up(w, off, 32);
      if (lane >= off) w += u;
    }
    wsum[lane] = w;
  }
  __syncthreads();

  const int base = (wave > 0) ? wsum[wave - 1] : 0;
  const int cum  = base + v;
  s[t] = cum;

  if (t == T_ - 1) {
    flen[b] = cum;
    out_flen[b] = (float)cum;
  }
  __syncthreads();

  // searchsorted(cum, p, side='right') = #entries <= p, clipped to T-1.
  #pragma unroll
  for (int k = 0; k < MAXF_ / T_; ++k) {
    const int p = t + k * T_;
    int lo = 0, hi = T_;
    while (lo < hi) {
      int mid = (lo + hi) >> 1;
      if (s[mid] <= p) lo = mid + 1;
      else             hi = mid;
    }
    int tk = (lo < T_ - 1) ? lo : (T_ - 1);
    tok[b * MAXF_ + p] = tk;  // coalesced along p
  }
}

// -----------------------------------------------------------------------------
// Kernel B: gather/expand. Block = (frame tile of 1024, channel group of 32,
// batch). Blocks entirely past frame_lengths (avg ~55% of the output) take a
// pure zero-store fast path. Otherwise: tok tile staged into LDS with async
// global->LDS b128 (ASYNCcnt), gathers of x hit L2 (x = 16 MiB << 192 MB L2),
// output written with 16B non-temporal stores (128 MiB streaming).
// -----------------------------------------------------------------------------
__global__ __launch_bounds__(TPB) void lr_gather_kernel(
    const float* __restrict__ x,    // [B][C][T]
    const int* __restrict__ tok,    // [B][MAXF]
    const int* __restrict__ flen,   // [B]
    float* __restrict__ out)        // [B][C][MAXF]
{
  __shared__ __align__(16) int s_tok[TP];

  const int b   = blockIdx.z;
  const int c0  = blockIdx.y * CC;
  const int p0  = blockIdx.x * TP;
  const int tid = threadIdx.x;

  const int fl = flen[b];  // scalar (uniform) load
  const int p  = p0 + tid * 4;

  float* orow = out + ((size_t)b * C_ + c0) * MAXF_ + p;

  if (p0 >= fl) {
    // Entire frame tile is past the regulated length: zeros only, no gathers.
    const v4f z = {0.0f, 0.0f, 0.0f, 0.0f};
    #pragma unroll 4
    for (int cc = 0; cc < CC; ++cc) {
      __builtin_nontemporal_store(z, (v4f*)orow);
      orow += MAXF_;
    }
    return;
  }

  const int* gsrc = tok + b * MAXF_ + p0 + tid * 4;  // 16B aligned

#if HAVE_ASYNC_LDS
  __builtin_amdgcn_global_load_async_to_lds_b128(
      (__attribute__((address_space(1))) v4i*)gsrc,
      (__attribute__((address_space(3))) v4i*)&s_tok[tid * 4],
      /*offset=*/0, /*cpol=*/0);
  __builtin_amdgcn_s_wait_asynccnt(0);
#else
  *(v4i*)&s_tok[tid * 4] = *(const v4i*)gsrc;
#endif
  __syncthreads();

  // Hoist token indices + masks for this thread's 4 frames.
  int  tk[4];
  bool m[4];
  #pragma unroll
  for (int j = 0; j < 4; ++j) {
    tk[j] = s_tok[tid * 4 + j];
    m[j]  = (p + j) < fl;
  }

  const float* xrow = x + ((size_t)b * C_ + c0) * T_;

  for (int cc = 0; cc < CC; ++cc) {
    // Prefetch next channel row of x (uniform address; OOB tail is dropped).
    __builtin_prefetch(xrow + T_, 0, 0);

    v4f v;
    v[0] = m[0] ? xrow[tk[0]] : 0.0f;
    v[1] = m[1] ? xrow[tk[1]] : 0.0f;
    v[2] = m[2] ? xrow[tk[2]] : 0.0f;
    v[3] = m[3] ? xrow[tk[3]] : 0.0f;

    __builtin_nontemporal_store(v, (v4f*)orow);  // streaming 128 MiB output

    xrow += T_;
    orow += MAXF_;
  }
}

// -----------------------------------------------------------------------------
// Launch
// -----------------------------------------------------------------------------
extern "C" void kernel_launch(void* const* d_in, const int* in_sizes, int n_in,
                              void* d_out, int out_size, void* d_ws, size_t ws_size,
                              hipStream_t stream) {
  const float* x        = (const float*)d_in[0];  // (16,256,1024) fp32
  const int*   duration = (const int*)d_in[1];    // (16,1024) int
  // d_in[2] = x_lengths: unused by the reference computation.

  float* out = (float*)d_out;                         // [B][C][MAXF] + [B] flen
  float* out_flen = out + (size_t)B_ * C_ * MAXF_;

  int* tok  = (int*)d_ws;          // [B][MAXF]  = 512 KiB
  int* flen = tok + B_ * MAXF_;    // [B]

  lr_scan_tok_kernel<<<B_, T_, 0, stream>>>(duration, tok, flen, out_flen);

  dim3 grid(MAXF_ / TP, C_ / CC, B_);  // (8, 8, 16) = 1024 blocks
  lr_gather_kernel<<<grid, TPB, 0, stream>>>(x, tok, flen, out);
}